// GCNModel_91233695301948
// MI455X (gfx1250) — compile-verified
//
#include <hip/hip_runtime.h>

typedef __attribute__((ext_vector_type(2))) float v2f;
typedef __attribute__((ext_vector_type(8))) float v8f;

#define TB 256

// ---------------- graph-invariant precompute ----------------

__global__ void k_fill(float* __restrict__ p, float v, int n) {
    int i = blockIdx.x * blockDim.x + threadIdx.x;
    if (i < n) p[i] = v;
}

__global__ void k_cvt_idx(const long long* __restrict__ i64, int* __restrict__ i32, int n) {
    int i = blockIdx.x * blockDim.x + threadIdx.x;
    if (i < n) i32[i] = (int)__builtin_nontemporal_load(&i64[i]);
}

template <typename IT>
__global__ void k_deg(const IT* __restrict__ dst, float* __restrict__ deg, int E) {
    int e = blockIdx.x * blockDim.x + threadIdx.x;
    if (e < E) atomicAdd(&deg[(int)__builtin_nontemporal_load(&dst[e])], 1.0f);
}

__global__ void k_rsqrt(float* __restrict__ p, int n) {
    int i = blockIdx.x * blockDim.x + threadIdx.x;
    if (i < n) p[i] = rsqrtf(p[i]);   // deg >= 1 always (self loop)
}

template <typename IT>
__global__ void k_norm(const IT* __restrict__ src, const IT* __restrict__ dst,
                       const float* __restrict__ dinv, float* __restrict__ norm, int E) {
    int e = blockIdx.x * blockDim.x + threadIdx.x;
    if (e < E) {
        int s = (int)__builtin_nontemporal_load(&src[e]);
        int d = (int)__builtin_nontemporal_load(&dst[e]);
        norm[e] = dinv[s] * dinv[d];
    }
}

// ---------------- dense lift: 16-row tile per wave, V_WMMA_F32_16X16X4_F32 ----------------
// A layout: lane m=lane&15 -> M=m; kh=lane>>4 selects K-half; vgpr v -> K = 4c + 2*kh + v.
// C/D layout: vgpr r, lanes 0-15 -> (M=r, N=lane), lanes 16-31 -> (M=r+8, N=lane-16).
// Loads unconditional (clamped 32-bit offsets); masking via selects -> EXEC all-ones at WMMA.

template <int K, int NOUT, bool HAS_BIAS, bool PADDED>
__global__ void k_gemm_wmma(const float* __restrict__ X, int ldx,
                            const float* __restrict__ W,
                            const float* __restrict__ bias,
                            float* __restrict__ Hout, int ldh,
                            int nTiles, int N) {
    int wv   = (blockIdx.x * blockDim.x + threadIdx.x) >> 5;
    int lane = threadIdx.x & 31;
    if (wv >= nTiles) return;                 // wave-uniform exit
    int row0 = wv << 4;
    int m    = lane & 15;
    int kh   = lane >> 4;                     // 0/1
    int arow = row0 + m;
    int crow = arow < N ? arow : N - 1;       // clamped, branch-free safe load
    int mc   = m < NOUT ? m : NOUT - 1;
    int xbase = crow * ldx;                   // 32-bit offsets throughout

    v8f acc = {};
    constexpr int NC = (K + 3) / 4;
#pragma unroll
    for (int c = 0; c < NC; ++c) {
        v2f a, b;
#pragma unroll
        for (int v = 0; v < 2; ++v) {
            int k   = (c << 2) + (kh << 1) + v;
            int kc  = k < K ? k : K - 1;
            float xv  = X[xbase + kc];                    // unconditional load
            float wvv = W[kc * NOUT + mc];                // unconditional load
            a[v] = (k < K) ? xv : 0.0f;                   // v_cndmask
            b[v] = (k < K && m < NOUT) ? wvv : 0.0f;      // v_cndmask
        }
        acc = __builtin_amdgcn_wmma_f32_16x16x4_f32(
            false, a, false, b, (short)0, acc, false, false);
    }

    if (m < NOUT) {                           // single divergence point after WMMA
        int base = (row0 + (kh << 3)) * ldh + m;
#pragma unroll
        for (int r = 0; r < 8; ++r) {
            float out = acc[r];
            if constexpr (HAS_BIAS) out += bias[mc];
            if constexpr (PADDED) {
                Hout[base + r * ldh] = out;               // buffer padded to nTiles*16 rows
            } else {
                if (row0 + r + (kh << 3) < N) Hout[base + r * ldh] = out;
            }
        }
    }
}

// ---------------- per-node: out = h * dinv^2 (self loop) + bias ----------------

__global__ void k_init_out(const float* __restrict__ H, const float* __restrict__ dinv,
                           const float* __restrict__ bias, float* __restrict__ O, int N) {
    int i = blockIdx.x * blockDim.x + threadIdx.x;
    if (i >= N) return;
    float di = dinv[i];
    float w  = di * di;
    int  r8  = i << 3;
    float4 h0 = *(const float4*)(H + r8);
    float2 h1 = *(const float2*)(H + r8 + 4);
    float4 o0; float2 o1;
    o0.x = h0.x * w + bias[0];
    o0.y = h0.y * w + bias[1];
    o0.z = h0.z * w + bias[2];
    o0.w = h0.w * w + bias[3];
    o1.x = h1.x * w + bias[4];
    o1.y = h1.y * w + bias[5];
    *(float4*)(O + r8)     = o0;
    *(float2*)(O + r8 + 4) = o1;
}

// ---------------- edge scatter: O[dst] += H[src] * norm ----------------

template <typename IT>
__global__ void k_scatter(const IT* __restrict__ src, const IT* __restrict__ dst,
                          const float* __restrict__ norm, const float* __restrict__ H,
                          float* __restrict__ O, int E) {
    int e = blockIdx.x * blockDim.x + threadIdx.x;
    if (e >= E) return;
    int s   = (int)__builtin_nontemporal_load(&src[e]);
    int d   = (int)__builtin_nontemporal_load(&dst[e]);
    float w = __builtin_nontemporal_load(&norm[e]);
    int so = s << 3, dofs = d << 3;                    // 32-bit offsets
    float4 h0 = *(const float4*)(H + so);              // hot, L2-resident
    float2 h1 = *(const float2*)(H + so + 4);
    float* op = O + dofs;
    atomicAdd(op + 0, h0.x * w);
    atomicAdd(op + 1, h0.y * w);
    atomicAdd(op + 2, h0.z * w);
    atomicAdd(op + 3, h0.w * w);
    atomicAdd(op + 4, h1.x * w);
    atomicAdd(op + 5, h1.y * w);
}

// ---------------- fused LayerNorm + residual + ReLU (in-place on X) ----------------

__global__ void k_post(float* __restrict__ X, const float* __restrict__ O,
                       const float* __restrict__ gamma, const float* __restrict__ beta, int N) {
    int i = blockIdx.x * blockDim.x + threadIdx.x;
    if (i >= N) return;
    int r8 = i << 3;
    float4 a0 = *(const float4*)(O + r8);
    float2 a1 = *(const float2*)(O + r8 + 4);
    float o[6] = {a0.x, a0.y, a0.z, a0.w, a1.x, a1.y};
    float4 x0 = *(const float4*)(X + r8);
    float2 x1 = *(const float2*)(X + r8 + 4);
    float xr[6] = {x0.x, x0.y, x0.z, x0.w, x1.x, x1.y};

    float mu = 0.0f;
#pragma unroll
    for (int f = 0; f < 6; ++f) mu += o[f];
    mu *= (1.0f / 6.0f);
    float var = 0.0f;
#pragma unroll
    for (int f = 0; f < 6; ++f) { float dd = o[f] - mu; var += dd * dd; }
    var *= (1.0f / 6.0f);
    float inv = rsqrtf(var + 1e-5f);

    float r[6];
#pragma unroll
    for (int f = 0; f < 6; ++f) {
        float v = (o[f] - mu) * inv * gamma[f] + beta[f] + xr[f];
        r[f] = v > 0.0f ? v : 0.0f;
    }
    *(float4*)(X + r8)     = make_float4(r[0], r[1], r[2], r[3]);
    *(float2*)(X + r8 + 4) = make_float2(r[4], r[5]);
}

// ---------------- driver ----------------

extern "C" void kernel_launch(void* const* d_in, const int* in_sizes, int n_in,
                              void* d_out, int out_size, void* d_ws, size_t ws_size,
                              hipStream_t stream) {
    (void)n_in; (void)out_size;
    const float*     node  = (const float*)d_in[0];
    const long long* edges = (const long long*)d_in[1];
    const float*     W1    = (const float*)d_in[2];
    const float*     b1    = (const float*)d_in[3];
    const float*     Wl    = (const float*)d_in[4];
    const float*     bl    = (const float*)d_in[5];
    const float*     gamma = (const float*)d_in[6];
    const float*     beta  = (const float*)d_in[7];
    const float*     Wfc   = (const float*)d_in[8];
    const float*     bfc   = (const float*)d_in[9];

    const int N     = in_sizes[0] / 3;
    const int E     = in_sizes[1] / 2;
    const int STEPS = in_sizes[4] / 36;

    const long long* src64 = edges;       // edges[0]
    const long long* dst64 = edges + E;   // edges[1]

    const int nTiles = (N + 15) >> 4;
    const int Npad   = nTiles * 16;       // padded rows: gemm loads/stores always in-bounds

    // workspace layout (256B aligned)
    char*  ws  = (char*)d_ws;
    size_t off = 0;
    auto alloc = [&](size_t bytes) -> void* {
        void* p = ws + off;
        off += (bytes + 255) & ~(size_t)255;
        return p;
    };
    float* dinv = (float*)alloc((size_t)N * 4);
    float* norm = (float*)alloc((size_t)E * 4);
    float* X    = (float*)alloc((size_t)Npad * 32);
    float* H    = (float*)alloc((size_t)Npad * 32);
    float* O    = (float*)alloc((size_t)Npad * 32);
    // optional int32 index copy (halves streamed index bytes per conv)
    size_t idxBytes = (size_t)2 * E * 4;
    bool   use32    = (off + idxBytes + 256) <= ws_size;
    int*   idx32    = use32 ? (int*)alloc(idxBytes) : nullptr;
    const int* src32 = idx32;
    const int* dst32 = idx32 ? idx32 + E : nullptr;

    const dim3 blk(TB);
    const dim3 gN((N + TB - 1) / TB);
    const dim3 gE((E + TB - 1) / TB);
    const dim3 g2E((2 * E + TB - 1) / TB);
    const dim3 gT((nTiles * 32 + TB - 1) / TB);   // one wave per 16-row tile

    // graph invariants: computed ONCE, reused across all 5 convs
    k_fill<<<gN, blk, 0, stream>>>(dinv, 1.0f, N);           // self-loop contributes 1
    if (use32) {
        k_cvt_idx<<<g2E, blk, 0, stream>>>(edges, idx32, 2 * E);
        k_deg<int><<<gE, blk, 0, stream>>>(dst32, dinv, E);
        k_rsqrt<<<gN, blk, 0, stream>>>(dinv, N);
        k_norm<int><<<gE, blk, 0, stream>>>(src32, dst32, dinv, norm, E);
    } else {
        k_deg<long long><<<gE, blk, 0, stream>>>(dst64, dinv, E);
        k_rsqrt<<<gN, blk, 0, stream>>>(dinv, N);
        k_norm<long long><<<gE, blk, 0, stream>>>(src64, dst64, dinv, norm, E);
    }

    auto scatter = [&](const float* Hs, float* Os) {
        if (use32) k_scatter<int><<<gE, blk, 0, stream>>>(src32, dst32, norm, Hs, Os, E);
        else       k_scatter<long long><<<gE, blk, 0, stream>>>(src64, dst64, norm, Hs, Os, E);
    };

    // conv0: X = D^-1/2 (A+I) D^-1/2 (node @ W1) + b1
    k_gemm_wmma<3, 6, false, true><<<gT, blk, 0, stream>>>(node, 3, W1, nullptr, H, 8, nTiles, N);
    k_init_out<<<gN, blk, 0, stream>>>(H, dinv, b1, X, N);
    scatter(H, X);

    // residual GCN blocks
    for (int i = 0; i < STEPS; ++i) {
        k_gemm_wmma<6, 6, false, true><<<gT, blk, 0, stream>>>(X, 8, Wl + 36 * i, nullptr, H, 8, nTiles, N);
        k_init_out<<<gN, blk, 0, stream>>>(H, dinv, bl + 6 * i, O, N);
        scatter(H, O);
        k_post<<<gN, blk, 0, stream>>>(X, O, gamma + 6 * i, beta + 6 * i, N);
    }

    // final fc: out = X @ Wfc + bfc  (direct to d_out, stride 3; row-guarded stores)
    k_gemm_wmma<6, 3, true, false><<<gT, blk, 0, stream>>>(X, 8, Wfc, bfc, (float*)d_out, 3, nTiles, N);
}